// SS2D_19258633355925
// MI455X (gfx1250) — compile-verified
//
#include <hip/hip_runtime.h>
#include <hip/hip_bf16.h>

typedef __attribute__((ext_vector_type(2))) float v2f;
typedef __attribute__((ext_vector_type(8))) float v8f;
typedef __attribute__((ext_vector_type(4))) unsigned int v4u;
typedef __attribute__((ext_vector_type(8))) int v8i;
typedef __attribute__((ext_vector_type(4))) int v4i;

#define BN   2
#define HH   128
#define WWD  128
#define CIN  96
#define DI   192
#define LL   16384
#define KD   4
#define NS   16
#define RD   6
#define BL   (BN * LL)
#define CDBL 38
#define CH   64            // scan steps per TDM chunk
#define NCH  (LL / CH)     // 256 chunks

__device__ __forceinline__ float silu_f(float x) {
    return x / (1.0f + __expf(-x));
}
__device__ __forceinline__ float softplus_f(float x) {
    return (x > 20.0f) ? x : __logf(1.0f + __expf(x));
}
__device__ __forceinline__ int map_transpose(int l) {
    // l = w*H + h  ->  spatial pos h*W + w   (H == W == 128)
    return ((l & 127) << 7) | (l >> 7);
}

// ---------------------------------------------------------------------------
// TDM 2D tile load: tile_dim0 = 38 elems (4B), tile_dim1 = 64 rows,
// row stride = stride_elems.  Descriptor layout per CDNA5 ISA.
// This toolchain exposes the 6-arg builtin (extra int32x8 group + cpol).
// ---------------------------------------------------------------------------
__device__ __forceinline__ void tdm_load_rows(unsigned lds_off, const float* gptr,
                                              unsigned stride_elems) {
    const unsigned long long ga = (unsigned long long)(size_t)gptr;
    v4u g0;
    g0.x = 1u;                                           // count=1, user D#
    g0.y = lds_off;                                      // lds_addr (bytes)
    g0.z = (unsigned)(ga & 0xffffffffull);               // global_addr[31:0]
    g0.w = (unsigned)((ga >> 32) & 0x01ffffffull)        // global_addr[56:32]
         | (2u << 30);                                   // type = 2 ("image")
    v8i g1;
    g1[0] = (int)(2u << 16);              // workgroup_mask=0, data_size=2 (4B)
    g1[1] = (int)((unsigned)CDBL << 16);  // barrier_addr=0 | tensor_dim0 lo16 = 38
    g1[2] = (int)(0x4000u << 16);         // tensor_dim0 hi16=0 | tensor_dim1 lo16 = 16384
    g1[3] = (int)((unsigned)CDBL << 16);  // tensor_dim1 hi16=0 | tile_dim0 = 38
    g1[4] = CH;                           // tile_dim1 = 64, tile_dim2 = 0
    g1[5] = (int)stride_elems;            // tensor_dim0_stride lo32
    g1[6] = 0;                            // stride0 hi16 | dim1_stride lo16
    g1[7] = 0;
    v4i g2 = {0, 0, 0, 0};
    v4i g3 = {0, 0, 0, 0};
    v8i g4 = {0, 0, 0, 0, 0, 0, 0, 0};
    __builtin_amdgcn_tensor_load_to_lds(g0, g1, g2, g3, g4, 0);
}

// ---------------------------------------------------------------------------
// Kernel 1: in_proj GEMM  (BL x 96) @ (96 x 384) -> xi (BL x 192), z = silu(..)
// ---------------------------------------------------------------------------
__global__ __launch_bounds__(128) void k_inproj(const float* __restrict__ x,
                                                const float* __restrict__ w,   // (384,96)
                                                float* __restrict__ xi,
                                                float* __restrict__ z) {
    const int lane = threadIdx.x & 31;
    const int wave = threadIdx.x >> 5;
    const int m0 = blockIdx.x * 16;
    const int n0 = (blockIdx.y * 4 + wave) * 16;
    const int h = lane >> 4;
    const int r = lane & 15;

    v8f acc = {};
    const float* arow = x + (size_t)(m0 + r) * CIN;
    const float* brow = w + (size_t)(n0 + r) * CIN;
    for (int kk = 0; kk < CIN; kk += 4) {
        v2f a = *(const v2f*)(arow + kk + 2 * h);
        v2f b = *(const v2f*)(brow + kk + 2 * h);
        acc = __builtin_amdgcn_wmma_f32_16x16x4_f32(false, a, false, b,
                                                    (short)0, acc, false, false);
    }
    for (int j = 0; j < 8; ++j) {
        const int row = m0 + j + 8 * h;
        const int col = n0 + r;
        const float v = acc[j];
        if (col < DI) {
            xi[(size_t)row * DI + col] = v;
        } else {
            z[(size_t)row * DI + (col - DI)] = silu_f(v);
        }
    }
}

// ---------------------------------------------------------------------------
// Kernel 2: depthwise 3x3 conv + bias + SiLU, layout (B, L, DI)
// ---------------------------------------------------------------------------
__global__ __launch_bounds__(256) void k_conv(const float* __restrict__ xi,
                                              const float* __restrict__ cw,
                                              const float* __restrict__ cb,
                                              float* __restrict__ xc) {
    const int idx = blockIdx.x * 256 + threadIdx.x;
    const int d = idx % DI;
    const int p = (idx / DI) % LL;
    const int b = idx / (DI * LL);
    const int hh = p >> 7;
    const int ww = p & 127;
    float acc = cb[d];
    #pragma unroll
    for (int dy = -1; dy <= 1; ++dy) {
        const int y = hh + dy;
        if (y < 0 || y >= HH) continue;
        #pragma unroll
        for (int dx = -1; dx <= 1; ++dx) {
            const int xw = ww + dx;
            if (xw < 0 || xw >= WWD) continue;
            acc += xi[((size_t)b * LL + ((y << 7) + xw)) * DI + d]
                 * cw[d * 9 + (dy + 1) * 3 + (dx + 1)];
        }
    }
    xc[idx] = silu_f(acc);
}

// ---------------------------------------------------------------------------
// Kernel 3: x_dbl GEMM per (b, k): (L x 192) @ (192 x 48pad) -> (b,k,L,38)
// ---------------------------------------------------------------------------
__global__ __launch_bounds__(96) void k_xdbl(const float* __restrict__ xc,
                                             const float* __restrict__ xpw,
                                             float* __restrict__ xdbl) {
    const int lane = threadIdx.x & 31;
    const int wave = threadIdx.x >> 5;
    const int m0 = blockIdx.x * 16;
    const int n0 = wave * 16;
    const int bk = blockIdx.y;
    const int b = bk >> 2;
    const int kdir = bk & 3;
    const int h = lane >> 4;
    const int r = lane & 15;

    const int c = n0 + r;
    const int cc = (c < CDBL) ? c : (CDBL - 1);
    const float msk = (c < CDBL) ? 1.0f : 0.0f;

    v8f acc = {};
    const float* arow = xc + ((size_t)b * LL + m0 + r) * DI;
    const float* brow = xpw + ((size_t)kdir * CDBL + cc) * DI;
    for (int kk = 0; kk < DI; kk += 4) {
        v2f a = *(const v2f*)(arow + kk + 2 * h);
        v2f bv = *(const v2f*)(brow + kk + 2 * h);
        bv.x *= msk; bv.y *= msk;
        acc = __builtin_amdgcn_wmma_f32_16x16x4_f32(false, a, false, bv,
                                                    (short)0, acc, false, false);
    }
    if (c < CDBL) {
        for (int j = 0; j < 8; ++j) {
            const int row = m0 + j + 8 * h;
            xdbl[((size_t)bk * LL + row) * CDBL + c] = acc[j];
        }
    }
}

// ---------------------------------------------------------------------------
// Kernel 4: selective scan with TDM double-buffered chunk prefetch.
// Block = 512 thr = 32 d x 16 n; one block per (d-chunk, b, direction-pair).
// Each 64-step chunk's x_dbl rows (fwd + bwd) arrive via tensor_load_to_lds
// as 2D tiles; wave 0 issues descriptors and gates on s_wait_tensorcnt.
// ---------------------------------------------------------------------------
__global__ __launch_bounds__(512) void k_scan(const float* __restrict__ xc,
                                              const float* __restrict__ xdbl,
                                              const float* __restrict__ alog,
                                              const float* __restrict__ dsv,
                                              const float* __restrict__ dtw,
                                              const float* __restrict__ dtb,
                                              float* __restrict__ yA,
                                              float* __restrict__ yB) {
    const int tid = threadIdx.x;
    const int n = tid & 15;
    const int dl = tid >> 4;
    const int chunk = blockIdx.x;
    const int b = blockIdx.y;
    const int pair = blockIdx.z;
    const int d = chunk * 32 + dl;
    const int kf = pair;
    const int kb = pair + 2;

    const float Af = -__expf(alog[((size_t)(kf * DI + d)) * NS + n]);
    const float Ab = -__expf(alog[((size_t)(kb * DI + d)) * NS + n]);
    const float Dsf = dsv[kf * DI + d];
    const float Dsb = dsv[kb * DI + d];
    const float dbf = dtb[kf * DI + d];
    const float dbb = dtb[kb * DI + d];
    float wf[RD], wb[RD];
    #pragma unroll
    for (int r = 0; r < RD; ++r) {
        wf[r] = dtw[((size_t)(kf * DI + d)) * RD + r];
        wb[r] = dtw[((size_t)(kb * DI + d)) * RD + r];
    }

    __shared__ float smF[2][CH * CDBL];   // fwd rows, double-buffered
    __shared__ float smB[2][CH * CDBL];   // bwd rows, double-buffered

    float hf = 0.0f, hb = 0.0f;
    float* __restrict__ Y = pair ? yB : yA;
    const float* xdb_f = xdbl + ((size_t)(b * KD + kf)) * LL * CDBL;
    const float* xdb_b = xdbl + ((size_t)(b * KD + kb)) * LL * CDBL;
    const unsigned strF = pair ? (unsigned)(128 * CDBL) : (unsigned)CDBL;

    // spatial row of the first tile row for chunk c:
    //  fwd: t0 = c*CH            bwd: lb0 = LL - (c+1)*CH (rows ascend with lb)
    auto row0f = [&](int c) -> int {
        const int t0 = c * CH;
        return pair ? (((t0 & 127) << 7) | (t0 >> 7)) : t0;
    };
    auto row0b = [&](int c) -> int {
        const int lb0 = LL - (c + 1) * CH;
        return pair ? (((lb0 & 127) << 7) | (lb0 >> 7)) : lb0;
    };

    if (tid == 0) {   // wave 0 issues chunk 0 (TDM ignores EXEC; wave-level issue)
        tdm_load_rows((unsigned)(size_t)&smF[0][0], xdb_f + (size_t)row0f(0) * CDBL, strF);
        tdm_load_rows((unsigned)(size_t)&smB[0][0], xdb_b + (size_t)row0b(0) * CDBL, strF);
    }

    for (int c = 0; c < NCH; ++c) {
        const int bp = c & 1;
        if (tid == 0) {
            if (c + 1 < NCH) {
                const int np = (c + 1) & 1;
                tdm_load_rows((unsigned)(size_t)&smF[np][0],
                              xdb_f + (size_t)row0f(c + 1) * CDBL, strF);
                tdm_load_rows((unsigned)(size_t)&smB[np][0],
                              xdb_b + (size_t)row0b(c + 1) * CDBL, strF);
                __builtin_amdgcn_s_wait_tensorcnt(2);   // chunk c's 2 tiles done
            } else {
                __builtin_amdgcn_s_wait_tensorcnt(0);
            }
        }
        __syncthreads();

        for (int i = 0; i < CH; ++i) {
            const int t = c * CH + i;
            const int lb = LL - 1 - t;
            const int pf = pair ? map_transpose(t) : t;
            const int pb = pair ? map_transpose(lb) : lb;
            const float* rf = &smF[bp][i * CDBL];
            const float* rb = &smB[bp][(CH - 1 - i) * CDBL];

            const float uf = xc[((size_t)b * LL + pf) * DI + d];
            const float ub = xc[((size_t)b * LL + pb) * DI + d];
            if (t + CH < LL) {  // prefetch next chunk's u row
                const int tn = t + CH;
                const int pn = pair ? map_transpose(tn) : tn;
                __builtin_prefetch(&xc[((size_t)b * LL + pn) * DI + d], 0, 3);
            }

            float dtf = dbf, dtbw = dbb;
            #pragma unroll
            for (int r = 0; r < RD; ++r) {
                dtf  += rf[r] * wf[r];
                dtbw += rb[r] * wb[r];
            }
            dtf = softplus_f(dtf);
            dtbw = softplus_f(dtbw);

            const float Bf = rf[RD + n], Cf = rf[RD + NS + n];
            const float Bb = rb[RD + n], Cb = rb[RD + NS + n];

            hf = __expf(dtf * Af) * hf + dtf * uf * Bf;
            hb = __expf(dtbw * Ab) * hb + dtbw * ub * Bb;

            float yf = hf * Cf;
            float yb = hb * Cb;
            #pragma unroll
            for (int m = 1; m < 16; m <<= 1) {
                yf += __shfl_xor(yf, m, 16);
                yb += __shfl_xor(yb, m, 16);
            }

            if (n == 0) {
                const float of = yf + Dsf * uf;
                const float ob = yb + Dsb * ub;
                const size_t iof = ((size_t)b * LL + pf) * DI + d;
                const size_t iob = ((size_t)b * LL + pb) * DI + d;
                if (t < LL / 2) { Y[iof] = of;  Y[iob] = ob;  }
                else            { Y[iof] += of; Y[iob] += ob; }
            }
        }
        __syncthreads();   // all reads of buffer bp done before chunk c+2 overwrites it
    }
}

// ---------------------------------------------------------------------------
// Kernel 5: merge + LayerNorm(192) + gate  ->  yg (B,L,DI)
// ---------------------------------------------------------------------------
__global__ __launch_bounds__(192) void k_ln(const float* __restrict__ yA,
                                            const float* __restrict__ yB,
                                            const float* __restrict__ z,
                                            const float* __restrict__ g,
                                            const float* __restrict__ bb,
                                            float* __restrict__ yg) {
    const int c = threadIdx.x;
    const size_t base = (size_t)blockIdx.x * DI;
    const float v = yA[base + c] + yB[base + c];

    float s = v, s2 = v * v;
    #pragma unroll
    for (int m = 16; m; m >>= 1) {
        s  += __shfl_xor(s, m, 32);
        s2 += __shfl_xor(s2, m, 32);
    }
    __shared__ float ss[6], ss2[6];
    const int wv = c >> 5;
    if ((c & 31) == 0) { ss[wv] = s; ss2[wv] = s2; }
    __syncthreads();
    if (c == 0) {
        float a = 0.0f, a2 = 0.0f;
        for (int i = 0; i < 6; ++i) { a += ss[i]; a2 += ss2[i]; }
        ss[0] = a; ss2[0] = a2;
    }
    __syncthreads();
    const float mean = ss[0] * (1.0f / DI);
    const float var  = ss2[0] * (1.0f / DI) - mean * mean;
    const float yn = (v - mean) * rsqrtf(var + 1e-5f) * g[c] + bb[c];
    yg[base + c] = yn * z[base + c];
}

// ---------------------------------------------------------------------------
// Kernel 6: out_proj GEMM  (BL x 192) @ (192 x 96) -> out
// ---------------------------------------------------------------------------
__global__ __launch_bounds__(192) void k_out(const float* __restrict__ yg,
                                             const float* __restrict__ opw,
                                             float* __restrict__ out) {
    const int lane = threadIdx.x & 31;
    const int wave = threadIdx.x >> 5;
    const int m0 = blockIdx.x * 16;
    const int n0 = wave * 16;
    const int h = lane >> 4;
    const int r = lane & 15;

    v8f acc = {};
    const float* arow = yg + (size_t)(m0 + r) * DI;
    const float* brow = opw + (size_t)(n0 + r) * DI;
    for (int kk = 0; kk < DI; kk += 4) {
        v2f a = *(const v2f*)(arow + kk + 2 * h);
        v2f b = *(const v2f*)(brow + kk + 2 * h);
        acc = __builtin_amdgcn_wmma_f32_16x16x4_f32(false, a, false, b,
                                                    (short)0, acc, false, false);
    }
    for (int j = 0; j < 8; ++j) {
        const int row = m0 + j + 8 * h;
        out[(size_t)row * CIN + (n0 + r)] = acc[j];
    }
}

// ---------------------------------------------------------------------------
extern "C" void kernel_launch(void* const* d_in, const int* in_sizes, int n_in,
                              void* d_out, int out_size, void* d_ws, size_t ws_size,
                              hipStream_t stream) {
    (void)in_sizes; (void)n_in; (void)out_size; (void)ws_size;
    const float* x    = (const float*)d_in[0];
    const float* ipw  = (const float*)d_in[1];
    const float* cw   = (const float*)d_in[2];
    const float* cb   = (const float*)d_in[3];
    const float* xpw  = (const float*)d_in[4];
    const float* dtw  = (const float*)d_in[5];
    const float* dtb  = (const float*)d_in[6];
    const float* alog = (const float*)d_in[7];
    const float* dsv  = (const float*)d_in[8];
    const float* lng  = (const float*)d_in[9];
    const float* lnb  = (const float*)d_in[10];
    const float* opw  = (const float*)d_in[11];
    float* out = (float*)d_out;

    float* ws = (float*)d_ws;
    const size_t SZ = (size_t)BL * DI;
    float* xi   = ws;                                  // reused as yg later
    float* zbuf = xi + SZ;
    float* xc   = zbuf + SZ;
    float* xdbl = xc + SZ;
    float* yA   = xdbl + (size_t)BN * KD * LL * CDBL;
    float* yB   = yA + SZ;
    float* yg   = xi;

    k_inproj<<<dim3(BL / 16, 384 / 64), 128, 0, stream>>>(x, ipw, xi, zbuf);
    k_conv<<<(BL * DI) / 256, 256, 0, stream>>>(xi, cw, cb, xc);
    k_xdbl<<<dim3(LL / 16, BN * KD), 96, 0, stream>>>(xc, xpw, xdbl);
    k_scan<<<dim3(6, BN, 2), 512, 0, stream>>>(xc, xdbl, alog, dsv, dtw, dtb, yA, yB);
    k_ln<<<BL, 192, 0, stream>>>(yA, yB, zbuf, lng, lnb, yg);
    k_out<<<BL / 16, 192, 0, stream>>>(yg, opw, out);
}